// GatedGCNNet2_57243324121155
// MI455X (gfx1250) — compile-verified
//
#include <hip/hip_runtime.h>

#define D_PAD 80
#define NCT 5   // 80/16 column tiles

typedef float v2f __attribute__((ext_vector_type(2)));
typedef float v8f __attribute__((ext_vector_type(8)));

// ---------------------------------------------------------------- utilities
__global__ void fill_kernel(float* __restrict__ p, long n, float v) {
  long t = (long)blockIdx.x * blockDim.x + threadIdx.x;
  if (t < n) p[t] = v;
}

// Shuffle a KxD (D=70, row-major) weight matrix into WMMA B-fragment order:
// for k-step ks, col-tile ct, lane: v2f = { W[kr][col], W[kr+1][col] },
// kr = ks*4 + (lane>=16 ? 2 : 0), col = ct*16 + (lane&15). Zero-padded.
__global__ void pack_wfrag_kernel(const float* __restrict__ W, int K, int ksteps,
                                  v2f* __restrict__ dst) {
  int t = blockIdx.x * blockDim.x + threadIdx.x;
  int total = ksteps * NCT * 32;
  if (t >= total) return;
  int lane = t & 31;
  int ct   = (t >> 5) % NCT;
  int ks   = t / (NCT * 32);
  int mlo  = lane & 15;
  int hi   = lane >> 4;
  int col  = ct * 16 + mlo;
  int kr   = ks * 4 + hi * 2;
  v2f v;
  v.x = (kr     < K && col < 70) ? W[(long)kr * 70 + col]       : 0.f;
  v.y = (kr + 1 < K && col < 70) ? W[(long)(kr + 1) * 70 + col] : 0.f;
  dst[t] = v;
}

__global__ void pad_vec_kernel(const float* __restrict__ s, float* __restrict__ d, int n) {
  int t = threadIdx.x;
  if (t < D_PAD) d[t] = (t < n) ? s[t] : 0.f;
}

// e[i][c] = edges_feat[i] * Wemb_e[c] + bemb_e[c]  (rank-1 embed), pad cols = 0
__global__ void edge_embed_kernel(const float* __restrict__ ef,
                                  const float* __restrict__ We,
                                  const float* __restrict__ be,
                                  float* __restrict__ e, long total) {
  long t = (long)blockIdx.x * blockDim.x + threadIdx.x;
  if (t >= total) return;
  int c = (int)(t % D_PAD);
  long i = t / D_PAD;
  e[t] = (c < 70) ? ef[i] * We[c] + be[c] : 0.f;
}

// ------------------------------------------------------------- WMMA GEMM
// One wave (32 threads) computes a 32-row x 80-col strip of  out = A@W + b
// as two 16-row WMMA tiles sharing the B fragments (double arithmetic
// intensity). The K loop is software-pipelined: next k-step's A (LDS) and B
// (global) fragments are fetched before the current step's 10 WMMAs issue,
// so waits cover loads issued a full iteration earlier.
// FUSE adds Dh[src[row]] + Eh[dst[row]] in the epilogue (edge gate GEMM).
template <bool FUSE>
__global__ void gemm_wmma_kernel(const float* __restrict__ A, int lda, int K, int ksteps,
                                 const v2f* __restrict__ Wf,
                                 const float* __restrict__ bias,
                                 float* __restrict__ out, long M,
                                 const float* __restrict__ Dh,
                                 const float* __restrict__ Eh,
                                 const int* __restrict__ src,
                                 const int* __restrict__ dst) {
  extern __shared__ float lds[];                 // 32 * Kpad floats
  const int  lane = threadIdx.x;
  const int  Kpad = ksteps * 4;
  const long row0 = (long)blockIdx.x * 32;

  for (int r = 0; r < 32; ++r) {
    long row = row0 + r;
    for (int c = lane; c < Kpad; c += 32)
      lds[r * Kpad + c] = (row < M && c < K) ? A[row * lda + c] : 0.f;
  }
  __syncthreads();

  const int mlo = lane & 15;
  const int hi  = lane >> 4;
  const float* a0p = &lds[mlo * Kpad + hi * 2];
  const float* a1p = &lds[(16 + mlo) * Kpad + hi * 2];

  v8f acc0[NCT] = {};
  v8f acc1[NCT] = {};

  // pipeline prologue: fragments for k-step 0
  v2f a0 = *(const v2f*)&a0p[0];
  v2f a1 = *(const v2f*)&a1p[0];
  v2f b[NCT];
#pragma unroll
  for (int ct = 0; ct < NCT; ++ct) b[ct] = Wf[ct * 32 + lane];

  for (int ks = 0; ks < ksteps - 1; ++ks) {
    // prefetch k-step ks+1
    v2f a0n = *(const v2f*)&a0p[(ks + 1) * 4];
    v2f a1n = *(const v2f*)&a1p[(ks + 1) * 4];
    v2f bn[NCT];
#pragma unroll
    for (int ct = 0; ct < NCT; ++ct) bn[ct] = Wf[((ks + 1) * NCT + ct) * 32 + lane];
    // compute k-step ks
#pragma unroll
    for (int ct = 0; ct < NCT; ++ct) {
      acc0[ct] = __builtin_amdgcn_wmma_f32_16x16x4_f32(
          false, a0, false, b[ct], (short)0, acc0[ct], false, false);
      acc1[ct] = __builtin_amdgcn_wmma_f32_16x16x4_f32(
          false, a1, false, b[ct], (short)0, acc1[ct], false, false);
    }
    a0 = a0n; a1 = a1n;
#pragma unroll
    for (int ct = 0; ct < NCT; ++ct) b[ct] = bn[ct];
  }
#pragma unroll
  for (int ct = 0; ct < NCT; ++ct) {
    acc0[ct] = __builtin_amdgcn_wmma_f32_16x16x4_f32(
        false, a0, false, b[ct], (short)0, acc0[ct], false, false);
    acc1[ct] = __builtin_amdgcn_wmma_f32_16x16x4_f32(
        false, a1, false, b[ct], (short)0, acc1[ct], false, false);
  }

  auto epilogue = [&](const v8f (&ac)[NCT], long tr0) {
#pragma unroll
    for (int i = 0; i < 8; ++i) {
      long row = tr0 + i + hi * 8;
      long s = 0, d = 0;
      if (FUSE) { s = src[row]; d = dst[row]; }
#pragma unroll
      for (int ct = 0; ct < NCT; ++ct) {
        int col = ct * 16 + mlo;
        float v = ac[ct][i] + bias[col];
        if (FUSE) v += Dh[s * D_PAD + col] + Eh[d * D_PAD + col];
        out[row * D_PAD + col] = v;
      }
    }
  };
  epilogue(acc0, row0);
  if (row0 + 16 < M) epilogue(acc1, row0 + 16);  // block-uniform guard
}

// --------------------------------------------------- message scatter (atomics)
__global__ void edge_scatter_kernel(const float* __restrict__ ex,
                                    const float* __restrict__ Bh,
                                    const int* __restrict__ src,
                                    const int* __restrict__ dst,
                                    float* __restrict__ num,
                                    float* __restrict__ den, long total) {
  long t = (long)blockIdx.x * blockDim.x + threadIdx.x;
  if (t >= total) return;
  long e = t / 70;
  int  c = (int)(t % 70);
  float g  = ex[e * D_PAD + c];
  float sg = 1.f / (1.f + __expf(-g));
  long s = src[e], d = dst[e];
  atomicAdd(&num[d * D_PAD + c], Bh[s * D_PAD + c] * sg);
  atomicAdd(&den[d * D_PAD + c], sg);
}

// -------------------------------- fused snorm + BN statistics (block partials)
__global__ void node_stats_kernel(const float* __restrict__ Ah,
                                  const float* __restrict__ num,
                                  const float* __restrict__ den,
                                  const float* __restrict__ snorm,
                                  float* __restrict__ hx,
                                  float* __restrict__ gsum, float* __restrict__ gsq,
                                  long M) {
  __shared__ float ssum[4][D_PAD];
  __shared__ float ssq[4][D_PAD];
  int c = threadIdx.x;            // 0..79
  int y = threadIdx.y;            // 0..3
  long base = (long)blockIdx.x * 256;
  float sum = 0.f, sq = 0.f;
  for (int it = 0; it < 64; ++it) {
    long row = base + y + it * 4;
    if (row < M) {
      long idx = row * D_PAD + c;
      float x = (Ah[idx] + num[idx] / (den[idx] + 1e-6f)) * snorm[row];
      hx[idx] = x;
      sum += x; sq += x * x;
    }
  }
  ssum[y][c] = sum; ssq[y][c] = sq;
  __syncthreads();
  if (y == 0) {
    atomicAdd(&gsum[c], ssum[0][c] + ssum[1][c] + ssum[2][c] + ssum[3][c]);
    atomicAdd(&gsq[c],  ssq[0][c]  + ssq[1][c]  + ssq[2][c]  + ssq[3][c]);
  }
}

__global__ void edge_stats_kernel(float* __restrict__ ex,
                                  const float* __restrict__ snorm_e,
                                  float* __restrict__ gsum, float* __restrict__ gsq,
                                  long M) {
  __shared__ float ssum[4][D_PAD];
  __shared__ float ssq[4][D_PAD];
  int c = threadIdx.x;
  int y = threadIdx.y;
  long base = (long)blockIdx.x * 256;
  float sum = 0.f, sq = 0.f;
  for (int it = 0; it < 64; ++it) {
    long row = base + y + it * 4;
    if (row < M) {
      long idx = row * D_PAD + c;
      float x = ex[idx] * snorm_e[row];
      ex[idx] = x;
      sum += x; sq += x * x;
    }
  }
  ssum[y][c] = sum; ssq[y][c] = sq;
  __syncthreads();
  if (y == 0) {
    atomicAdd(&gsum[c], ssum[0][c] + ssum[1][c] + ssum[2][c] + ssum[3][c]);
    atomicAdd(&gsq[c],  ssq[0][c]  + ssq[1][c]  + ssq[2][c]  + ssq[3][c]);
  }
}

__global__ void bn_finalize_kernel(const float* __restrict__ gsum,
                                   const float* __restrict__ gsq, float invM,
                                   float* __restrict__ mean, float* __restrict__ inv) {
  int c = threadIdx.x;
  if (c < D_PAD) {
    float m = gsum[c] * invM;
    float v = gsq[c] * invM - m * m;
    mean[c] = m;
    inv[c]  = rsqrtf(fmaxf(v, 0.f) + 1e-5f);
  }
}

// io += relu( (x - mean) * inv * gamma + beta )   (residual)
__global__ void bn_apply_res_kernel(float* __restrict__ io, const float* __restrict__ x,
                                    const float* __restrict__ mean,
                                    const float* __restrict__ inv,
                                    const float* __restrict__ gamma,
                                    const float* __restrict__ beta, long total) {
  long t = (long)blockIdx.x * blockDim.x + threadIdx.x;
  if (t >= total) return;
  int c = (int)(t % D_PAD);
  float y = (x[t] - mean[c]) * inv[c] * gamma[c] + beta[c];
  io[t] += fmaxf(y, 0.f);
}

// ------------------------------------------------------------- readout
__global__ void hg_accum_kernel(const float* __restrict__ h, const int* __restrict__ gid,
                                float* __restrict__ hg, long total) {
  long t = (long)blockIdx.x * blockDim.x + threadIdx.x;
  if (t >= total) return;
  long row = t / D_PAD;
  int  c   = (int)(t % D_PAD);
  atomicAdd(&hg[(long)gid[row] * D_PAD + c], h[t]);
}

__global__ void cnt_accum_kernel(const int* __restrict__ gid, float* __restrict__ cnt, int N) {
  int t = blockIdx.x * blockDim.x + threadIdx.x;
  if (t < N) atomicAdd(&cnt[gid[t]], 1.f);
}

__global__ void readout_kernel(const float* __restrict__ hg, const float* __restrict__ cnt,
                               const float* __restrict__ W0, const float* __restrict__ b0,
                               const float* __restrict__ W1, const float* __restrict__ b1,
                               const float* __restrict__ W2, const float* __restrict__ b2,
                               float* __restrict__ out) {
  __shared__ float x0[128 * 35];
  __shared__ float x1[128 * 17];
  int tid = threadIdx.x;
  for (int idx = tid; idx < 128 * 35; idx += 256) {
    int g = idx / 35, o = idx % 35;
    float invc = 1.f / fmaxf(cnt[g], 1.f);
    float s = b0[o];
    for (int c = 0; c < 70; ++c) s += hg[g * D_PAD + c] * invc * W0[c * 35 + o];
    x0[idx] = fmaxf(s, 0.f);
  }
  __syncthreads();
  for (int idx = tid; idx < 128 * 17; idx += 256) {
    int g = idx / 17, o = idx % 17;
    float s = b1[o];
    for (int c = 0; c < 35; ++c) s += x0[g * 35 + c] * W1[c * 17 + o];
    x1[idx] = fmaxf(s, 0.f);
  }
  __syncthreads();
  for (int idx = tid; idx < 128 * 10; idx += 256) {
    int g = idx / 10, o = idx % 10;
    float s = b2[o];
    for (int c = 0; c < 17; ++c) s += x1[g * 17 + c] * W2[c * 10 + o];
    out[idx] = s;
  }
}

// ================================================================= launch
extern "C" void kernel_launch(void* const* d_in, const int* in_sizes, int n_in,
                              void* d_out, int out_size, void* d_ws, size_t ws_size,
                              hipStream_t stream) {
  (void)in_sizes; (void)n_in; (void)out_size;
  const int N = 50000, E = 800000, G = 128, L = 4;
  const int KS_EMB = 37;  // ceil(146/4)
  const int KS_D   = 20;  // 80/4

  const float* nodes_feat = (const float*)d_in[0];
  const float* edges_feat = (const float*)d_in[1];
  const float* snorm_n    = (const float*)d_in[2];
  const float* snorm_e    = (const float*)d_in[3];
  const int*   src        = (const int*)d_in[4];
  const int*   dst        = (const int*)d_in[5];
  const int*   ngid       = (const int*)d_in[6];
  const float* Wemb_h = (const float*)d_in[7];
  const float* bemb_h = (const float*)d_in[8];
  const float* Wemb_e = (const float*)d_in[9];
  const float* bemb_e = (const float*)d_in[10];
  const float* Wmat[5] = { (const float*)d_in[11], (const float*)d_in[13],
                           (const float*)d_in[15], (const float*)d_in[17],
                           (const float*)d_in[19] };
  const float* bmat[5] = { (const float*)d_in[12], (const float*)d_in[14],
                           (const float*)d_in[16], (const float*)d_in[18],
                           (const float*)d_in[20] };
  const float* bn_gh = (const float*)d_in[21];
  const float* bn_bh = (const float*)d_in[22];
  const float* bn_ge = (const float*)d_in[23];
  const float* bn_be = (const float*)d_in[24];
  const float* W0 = (const float*)d_in[25]; const float* b0 = (const float*)d_in[26];
  const float* W1 = (const float*)d_in[27]; const float* b1 = (const float*)d_in[28];
  const float* W2 = (const float*)d_in[29]; const float* b2 = (const float*)d_in[30];
  float* out = (float*)d_out;

  // ---- workspace layout
  float* wsf = (float*)d_ws;
  size_t o = 0;
  auto alloc = [&](size_t n) { float* p = wsf + o; o += n; return p; };
  const size_t ND = (size_t)N * D_PAD, ED = (size_t)E * D_PAD;
  float* h   = alloc(ND);
  float* hA  = alloc(ND);
  float* hB  = alloc(ND);
  float* hD  = alloc(ND);
  float* hE  = alloc(ND);
  float* num = alloc(ND);
  float* den = alloc(ND);
  float* hx  = alloc(ND);
  float* e   = alloc(ED);
  float* ex  = alloc(ED);
  const size_t FRAG_EMB = (size_t)KS_EMB * NCT * 32;  // v2f count
  const size_t FRAG_D   = (size_t)KS_D   * NCT * 32;
  v2f* frag_emb = (v2f*)alloc(FRAG_EMB * 2);
  v2f* frag_lay = (v2f*)alloc((size_t)L * 5 * FRAG_D * 2);
  float* bias_pad = alloc((size_t)(1 + L * 5) * D_PAD);
  float* bng_h = alloc((size_t)L * D_PAD);
  float* bnb_h = alloc((size_t)L * D_PAD);
  float* bng_e = alloc((size_t)L * D_PAD);
  float* bnb_e = alloc((size_t)L * D_PAD);
  float* gsum  = alloc(D_PAD);
  float* gsq   = alloc(D_PAD);
  float* meanb = alloc(D_PAD);
  float* invb  = alloc(D_PAD);
  float* hg    = alloc((size_t)G * D_PAD);
  float* cnt   = alloc(G);
  if (ws_size < o * sizeof(float)) return;  // workspace too small: bail deterministically

  // ---- pack weights / biases / bn params into padded WMMA-friendly layouts
  pack_wfrag_kernel<<<(int)((FRAG_EMB + 255) / 256), 256, 0, stream>>>(Wemb_h, 146, KS_EMB, frag_emb);
  for (int l = 0; l < L; ++l)
    for (int m = 0; m < 5; ++m)
      pack_wfrag_kernel<<<(int)((FRAG_D + 255) / 256), 256, 0, stream>>>(
          Wmat[m] + (size_t)l * 70 * 70, 70, KS_D, frag_lay + ((size_t)l * 5 + m) * FRAG_D);
  pad_vec_kernel<<<1, 128, 0, stream>>>(bemb_h, bias_pad, 70);
  for (int l = 0; l < L; ++l)
    for (int m = 0; m < 5; ++m)
      pad_vec_kernel<<<1, 128, 0, stream>>>(bmat[m] + (size_t)l * 70,
                                            bias_pad + (size_t)(1 + l * 5 + m) * D_PAD, 70);
  for (int l = 0; l < L; ++l) {
    pad_vec_kernel<<<1, 128, 0, stream>>>(bn_gh + l * 70, bng_h + l * D_PAD, 70);
    pad_vec_kernel<<<1, 128, 0, stream>>>(bn_bh + l * 70, bnb_h + l * D_PAD, 70);
    pad_vec_kernel<<<1, 128, 0, stream>>>(bn_ge + l * 70, bng_e + l * D_PAD, 70);
    pad_vec_kernel<<<1, 128, 0, stream>>>(bn_be + l * 70, bnb_e + l * D_PAD, 70);
  }

  // ---- embeddings
  const size_t sh_emb = (size_t)32 * KS_EMB * 4 * sizeof(float);
  gemm_wmma_kernel<false><<<(N + 31) / 32, 32, sh_emb, stream>>>(
      nodes_feat, 146, 146, KS_EMB, frag_emb, bias_pad, h, (long)N,
      nullptr, nullptr, nullptr, nullptr);
  edge_embed_kernel<<<(int)((ED + 255) / 256), 256, 0, stream>>>(edges_feat, Wemb_e, bemb_e, e, (long)ED);

  // ---- GatedGCN layers
  const size_t sh_d = (size_t)32 * D_PAD * sizeof(float);
  for (int l = 0; l < L; ++l) {
    const v2f* fA = frag_lay + ((size_t)l * 5 + 0) * FRAG_D;
    const v2f* fB = frag_lay + ((size_t)l * 5 + 1) * FRAG_D;
    const v2f* fC = frag_lay + ((size_t)l * 5 + 2) * FRAG_D;
    const v2f* fD = frag_lay + ((size_t)l * 5 + 3) * FRAG_D;
    const v2f* fE = frag_lay + ((size_t)l * 5 + 4) * FRAG_D;
    const float* bpA = bias_pad + (size_t)(1 + l * 5 + 0) * D_PAD;
    const float* bpB = bias_pad + (size_t)(1 + l * 5 + 1) * D_PAD;
    const float* bpC = bias_pad + (size_t)(1 + l * 5 + 2) * D_PAD;
    const float* bpD = bias_pad + (size_t)(1 + l * 5 + 3) * D_PAD;
    const float* bpE = bias_pad + (size_t)(1 + l * 5 + 4) * D_PAD;

    gemm_wmma_kernel<false><<<(N + 31) / 32, 32, sh_d, stream>>>(h, D_PAD, D_PAD, KS_D, fA, bpA, hA, (long)N, nullptr, nullptr, nullptr, nullptr);
    gemm_wmma_kernel<false><<<(N + 31) / 32, 32, sh_d, stream>>>(h, D_PAD, D_PAD, KS_D, fB, bpB, hB, (long)N, nullptr, nullptr, nullptr, nullptr);
    gemm_wmma_kernel<false><<<(N + 31) / 32, 32, sh_d, stream>>>(h, D_PAD, D_PAD, KS_D, fD, bpD, hD, (long)N, nullptr, nullptr, nullptr, nullptr);
    gemm_wmma_kernel<false><<<(N + 31) / 32, 32, sh_d, stream>>>(h, D_PAD, D_PAD, KS_D, fE, bpE, hE, (long)N, nullptr, nullptr, nullptr, nullptr);
    // ex = e@WC + bC + Dh[src] + Eh[dst]
    gemm_wmma_kernel<true><<<E / 32, 32, sh_d, stream>>>(e, D_PAD, D_PAD, KS_D, fC, bpC, ex, (long)E, hD, hE, src, dst);

    fill_kernel<<<(int)((ND + 255) / 256), 256, 0, stream>>>(num, (long)ND, 0.f);
    fill_kernel<<<(int)((ND + 255) / 256), 256, 0, stream>>>(den, (long)ND, 0.f);
    long etot = (long)E * 70;
    edge_scatter_kernel<<<(int)((etot + 255) / 256), 256, 0, stream>>>(ex, hB, src, dst, num, den, etot);

    // node update: x = (Ah + num/(den+eps))*snorm_n ; BN ; relu ; residual
    fill_kernel<<<1, 128, 0, stream>>>(gsum, D_PAD, 0.f);
    fill_kernel<<<1, 128, 0, stream>>>(gsq, D_PAD, 0.f);
    node_stats_kernel<<<(N + 255) / 256, dim3(D_PAD, 4), 0, stream>>>(hA, num, den, snorm_n, hx, gsum, gsq, (long)N);
    bn_finalize_kernel<<<1, D_PAD, 0, stream>>>(gsum, gsq, 1.f / (float)N, meanb, invb);
    bn_apply_res_kernel<<<(int)((ND + 255) / 256), 256, 0, stream>>>(
        h, hx, meanb, invb, bng_h + l * D_PAD, bnb_h + l * D_PAD, (long)ND);

    // edge update: y = ex*snorm_e ; BN ; relu ; residual
    fill_kernel<<<1, 128, 0, stream>>>(gsum, D_PAD, 0.f);
    fill_kernel<<<1, 128, 0, stream>>>(gsq, D_PAD, 0.f);
    edge_stats_kernel<<<(E + 255) / 256, dim3(D_PAD, 4), 0, stream>>>(ex, snorm_e, gsum, gsq, (long)E);
    bn_finalize_kernel<<<1, D_PAD, 0, stream>>>(gsum, gsq, 1.f / (float)E, meanb, invb);
    bn_apply_res_kernel<<<(int)((ED + 255) / 256), 256, 0, stream>>>(
        e, ex, meanb, invb, bng_e + l * D_PAD, bnb_e + l * D_PAD, (long)ED);
  }

  // ---- readout
  fill_kernel<<<(int)(((size_t)G * D_PAD + 255) / 256), 256, 0, stream>>>(hg, (long)G * D_PAD, 0.f);
  fill_kernel<<<1, 128, 0, stream>>>(cnt, G, 0.f);
  hg_accum_kernel<<<(int)((ND + 255) / 256), 256, 0, stream>>>(h, ngid, hg, (long)ND);
  cnt_accum_kernel<<<(N + 255) / 256, 256, 0, stream>>>(ngid, cnt, N);
  readout_kernel<<<1, 256, 0, stream>>>(hg, cnt, W0, b0, W1, b1, W2, b2, out);
}